// VolumeAttentionLayer_82437602279998
// MI455X (gfx1250) — compile-verified
//
#include <hip/hip_runtime.h>

typedef __attribute__((ext_vector_type(2))) float v2f;
typedef __attribute__((ext_vector_type(8))) float v8f;

#define B_   2
#define C_   128
#define D_   48
#define H_   64
#define W_   128
#define NH   8
#define HW_  (H_ * W_)   // 8192
#define PIX  16

// LDS pool offsets (in floats)
static constexpr int CTX_OFF = 0;       // [128][16]      ctx -> ctx_ln
static constexpr int LNG_OFF = 2048;    // [128]          ln_g cache
static constexpr int LNB_OFF = 2176;    // [128]          ln_b cache
static constexpr int BO_OFF  = 2304;    // [128]          bo cache
static constexpr int K_OFF   = 2432;    // [128][16]      k
static constexpr int V_OFF   = 4480;    // [128][16]      v
static constexpr int SB_OFF  = 6528;    // [p*8+h]        bq.k bias
static constexpr int A2_OFF  = 6656;    // [p*8+h]        sum g*wk
static constexpr int A3_OFF  = 6784;    // [p*8+h]        sum b*wk
static constexpr int A2P_OFF = 6912;    // [8][32]        partials
static constexpr int A3P_OFF = 7168;    // [8][32]
static constexpr int WKG_OFF = 7424;    // [c][p][h] = 16384 floats (64KB); reused as U [oc][p][h]
static constexpr int U_OFF   = WKG_OFF;
static constexpr int SIM_OFF = 23808;   // [d][p][h] = 6144 floats (24KB); sim -> attn
static constexpr int SMEM_FLOATS = 29952; // ~117KB

__global__ __launch_bounds__(256, 1)
void vol_attn_kernel(const float* __restrict__ volume,
                     const float* __restrict__ feats,
                     const float* __restrict__ ln_g,  const float* __restrict__ ln_b,
                     const float* __restrict__ lnc_g, const float* __restrict__ lnc_b,
                     const float* __restrict__ Wq,    const float* __restrict__ bq,
                     const float* __restrict__ Wkv,   const float* __restrict__ bkv,
                     const float* __restrict__ Wo,    const float* __restrict__ bo,
                     float* __restrict__ out, float* __restrict__ attn_out)
{
    __shared__ __align__(16) float smem[SMEM_FLOATS];

    const int t    = threadIdx.x;
    const int lane = t & 31;
    const int wave = t >> 5;

    const int bid = blockIdx.x;           // b*512 + h*8 + wtile
    const int wt  = bid & 7;
    const int hh_ = (bid >> 3) & 63;
    const int bb  = bid >> 9;
    const int w0  = wt * PIX;

    // ---------------- Phase A: cache params + load feats tile ----------------
    if (t < 128) {
        smem[LNG_OFF + t] = ln_g[t];
        smem[LNB_OFF + t] = ln_b[t];
        smem[BO_OFF  + t] = bo[t];
    }
    for (int i = t; i < 128 * PIX; i += 256) {
        int c = i >> 4, p = i & 15;
        smem[CTX_OFF + i] = feats[((size_t)(bb * C_ + c) * H_ + hh_) * W_ + w0 + p];
    }
    __syncthreads();

    // ---------------- Phase B: LayerNorm of ctx (per pixel, over c) ----------
    if (t < PIX) {
        float s1 = 0.f, s2 = 0.f;
        for (int c = 0; c < 128; ++c) {
            float x = smem[CTX_OFF + c * 16 + t];
            s1 += x; s2 = fmaf(x, x, s2);
        }
        float mu  = s1 * (1.f / 128.f);
        float var = s2 * (1.f / 128.f) - mu * mu;
        float rs  = rsqrtf(var + 1e-6f);
        for (int c = 0; c < 128; ++c) {
            float x = smem[CTX_OFF + c * 16 + t];
            smem[CTX_OFF + c * 16 + t] = (x - mu) * rs * lnc_g[c] + lnc_b[c];
        }
    }
    __syncthreads();

    // ---------------- Phase C: KV[256,16] = Wkv @ ctx_ln  (WMMA f32 16x16x4) --
    {
        const int n     = lane & 15;
        const int khalf = (lane >> 4) * 2;
        for (int mt = wave; mt < 16; mt += 8) {
            v8f acc = {0.f,0.f,0.f,0.f,0.f,0.f,0.f,0.f};
            const int m = mt * 16 + (lane & 15);
            for (int kk = 0; kk < 32; ++kk) {
                const int kb = kk * 4 + khalf;
                v2f a, b;
                a.x = Wkv[(size_t)m * 128 + kb];
                a.y = Wkv[(size_t)m * 128 + kb + 1];
                b.x = smem[CTX_OFF + kb * 16 + n];
                b.y = smem[CTX_OFF + (kb + 1) * 16 + n];
                acc = __builtin_amdgcn_wmma_f32_16x16x4_f32(false, a, false, b,
                                                            (short)0, acc, false, false);
            }
            const int rbase = mt * 16 + ((lane >> 4) ? 8 : 0);
#pragma unroll
            for (int r = 0; r < 8; ++r) {
                const int row = rbase + r;
                const float val = acc[r] + bkv[row];
                if (row < 128) smem[K_OFF + row * 16 + n] = val;
                else           smem[V_OFF + (row - 128) * 16 + n] = val;
            }
        }
    }
    __syncthreads();

    // ---------------- Phase D: simbias[p][h] = bq_h . k_h ---------------------
    if (t < 128) {
        const int hh = t >> 4, p = t & 15;
        float s = 0.f;
        for (int j = 0; j < 16; ++j)
            s = fmaf(bq[hh * 16 + j], smem[K_OFF + (hh * 16 + j) * 16 + p], s);
        smem[SB_OFF + p * 8 + hh] = s;
    }

    // ---------------- Phase E: WK_h[128,16] = Wq_h^T @ K_h (WMMA, wave=head) --
    {
        const int hd    = wave;
        const int n     = lane & 15;
        const int khalf = (lane >> 4) * 2;
        float a2 = 0.f, a3 = 0.f;
        for (int mc = 0; mc < 8; ++mc) {
            v8f acc = {0.f,0.f,0.f,0.f,0.f,0.f,0.f,0.f};
            const int m = mc * 16 + (lane & 15);
            for (int ks = 0; ks < 4; ++ks) {
                const int jj = ks * 4 + khalf;
                v2f a, b;
                a.x = Wq[(size_t)(hd * 16 + jj) * 128 + m];
                a.y = Wq[(size_t)(hd * 16 + jj + 1) * 128 + m];
                b.x = smem[K_OFF + (hd * 16 + jj) * 16 + n];
                b.y = smem[K_OFF + (hd * 16 + jj + 1) * 16 + n];
                acc = __builtin_amdgcn_wmma_f32_16x16x4_f32(false, a, false, b,
                                                            (short)0, acc, false, false);
            }
            const int rbase = mc * 16 + ((lane >> 4) ? 8 : 0);
#pragma unroll
            for (int r = 0; r < 8; ++r) {
                const int c  = rbase + r;
                const float wk  = acc[r];
                const float wkg = smem[LNG_OFF + c] * wk;
                smem[WKG_OFF + (c * 16 + n) * 8 + hd] = wkg;
                a2 += wkg;
                a3 = fmaf(smem[LNB_OFF + c], wk, a3);
            }
        }
        smem[A2P_OFF + hd * 32 + lane] = a2;
        smem[A3P_OFF + hd * 32 + lane] = a3;
    }
    __syncthreads();
    if (t < 128) {
        const int hh = t >> 4, p = t & 15;
        smem[A2_OFF + p * 8 + hh] = smem[A2P_OFF + hh * 32 + p] + smem[A2P_OFF + hh * 32 + p + 16];
        smem[A3_OFF + p * 8 + hh] = smem[A3P_OFF + hh * 32 + p] + smem[A3P_OFF + hh * 32 + p + 16];
    }
    __syncthreads();

    // ---------------- Phase F: stream volume once: LN stats + head dots -------
    {
        const int p     = t & 15;
        const int dslot = t >> 4;
        const float scale = 0.25f;   // 16^-0.5
        for (int d = dslot; d < D_; d += 16) {
            float s1 = 0.f, s2 = 0.f;
            float acc0=0.f,acc1=0.f,acc2=0.f,acc3=0.f,acc4=0.f,acc5=0.f,acc6=0.f,acc7=0.f;
            const size_t vb = ((size_t)bb * C_ * D_ + (size_t)d) * HW_ + (size_t)hh_ * W_ + w0 + p;
#pragma unroll 4
            for (int c = 0; c < 128; ++c) {
                const float x = volume[vb + (size_t)c * (D_ * HW_)];
                s1 += x; s2 = fmaf(x, x, s2);
                const float4* wp = (const float4*)&smem[WKG_OFF + (c * 16 + p) * 8];
                const float4 wA = wp[0], wB = wp[1];
                acc0 = fmaf(x, wA.x, acc0); acc1 = fmaf(x, wA.y, acc1);
                acc2 = fmaf(x, wA.z, acc2); acc3 = fmaf(x, wA.w, acc3);
                acc4 = fmaf(x, wB.x, acc4); acc5 = fmaf(x, wB.y, acc5);
                acc6 = fmaf(x, wB.z, acc6); acc7 = fmaf(x, wB.w, acc7);
            }
            const float mu  = s1 * (1.f / 128.f);
            const float var = s2 * (1.f / 128.f) - mu * mu;
            const float rs  = rsqrtf(var + 1e-6f);
            float* sp = &smem[SIM_OFF + (d * 16 + p) * 8];
            const float acc[8] = {acc0,acc1,acc2,acc3,acc4,acc5,acc6,acc7};
#pragma unroll
            for (int hh = 0; hh < 8; ++hh) {
                sp[hh] = scale * (rs * (acc[hh] - mu * smem[A2_OFF + p * 8 + hh])
                                  + smem[A3_OFF + p * 8 + hh] + smem[SB_OFF + p * 8 + hh]);
            }
        }
    }
    __syncthreads();

    // ---------------- Phase G: softmax over d per (pixel, head) --------------
    if (t < 128) {
        const int p = t >> 3, hh = t & 7;
        float m = -1e30f;
        for (int d = 0; d < D_; ++d)
            m = fmaxf(m, smem[SIM_OFF + (d * 16 + p) * 8 + hh]);
        float s = 0.f;
        for (int d = 0; d < D_; ++d) {
            const float e = __expf(smem[SIM_OFF + (d * 16 + p) * 8 + hh] - m);
            smem[SIM_OFF + (d * 16 + p) * 8 + hh] = e;
            s += e;
        }
        const float inv = 1.f / s;
        for (int d = 0; d < D_; ++d)
            smem[SIM_OFF + (d * 16 + p) * 8 + hh] *= inv;
    }
    __syncthreads();

    // ---------------- Phase G2: write attn_out [b,n,d,h,w] (coalesced in w) --
    for (int i = t; i < NH * D_ * PIX; i += 256) {
        const int p = i & 15;
        const int rest = i >> 4;
        const int d = rest % D_;
        const int n = rest / D_;
        attn_out[((((size_t)bb * NH + n) * D_ + d) * H_ + hh_) * W_ + w0 + p] =
            smem[SIM_OFF + (d * 16 + p) * 8 + n];
    }

    // ---------------- Phase H: U_h[128,16] = Wo_h @ V_h (WMMA, reuse WKG LDS) -
    {
        const int hd    = wave;
        const int n     = lane & 15;
        const int khalf = (lane >> 4) * 2;
        for (int mc = 0; mc < 8; ++mc) {
            v8f acc = {0.f,0.f,0.f,0.f,0.f,0.f,0.f,0.f};
            const int m = mc * 16 + (lane & 15);
            for (int ks = 0; ks < 4; ++ks) {
                const int jj = ks * 4 + khalf;
                v2f a, b;
                a.x = Wo[(size_t)m * 128 + hd * 16 + jj];
                a.y = Wo[(size_t)m * 128 + hd * 16 + jj + 1];
                b.x = smem[V_OFF + (hd * 16 + jj) * 16 + n];
                b.y = smem[V_OFF + (hd * 16 + jj + 1) * 16 + n];
                acc = __builtin_amdgcn_wmma_f32_16x16x4_f32(false, a, false, b,
                                                            (short)0, acc, false, false);
            }
            const int rbase = mc * 16 + ((lane >> 4) ? 8 : 0);
#pragma unroll
            for (int r = 0; r < 8; ++r)
                smem[U_OFF + ((rbase + r) * 16 + n) * 8 + hd] = acc[r];
        }
    }
    __syncthreads();

    // ---------------- Phase I: O[d,oc] = sum_h attn*u + bo; write b,c,d,h,w --
    {
        const int p = t & 15;
        const int g = t >> 4;
        for (int idx = g; idx < D_ * 128; idx += 16) {
            const int oc = idx & 127;
            const int d  = idx >> 7;
            const float4* up = (const float4*)&smem[U_OFF + (oc * 16 + p) * 8];
            const float4* ap = (const float4*)&smem[SIM_OFF + (d * 16 + p) * 8];
            const float4 u0 = up[0], u1 = up[1];
            const float4 a0 = ap[0], a1 = ap[1];
            float o = smem[BO_OFF + oc];
            o = fmaf(a0.x, u0.x, o); o = fmaf(a0.y, u0.y, o);
            o = fmaf(a0.z, u0.z, o); o = fmaf(a0.w, u0.w, o);
            o = fmaf(a1.x, u1.x, o); o = fmaf(a1.y, u1.y, o);
            o = fmaf(a1.z, u1.z, o); o = fmaf(a1.w, u1.w, o);
            out[(((size_t)(bb * C_ + oc) * D_ + d) * H_ + hh_) * W_ + w0 + p] = o;
        }
    }
}

extern "C" void kernel_launch(void* const* d_in, const int* in_sizes, int n_in,
                              void* d_out, int out_size, void* d_ws, size_t ws_size,
                              hipStream_t stream) {
    const float* volume = (const float*)d_in[0];
    const float* feats  = (const float*)d_in[1];
    const float* ln_g   = (const float*)d_in[2];
    const float* ln_b   = (const float*)d_in[3];
    const float* lnc_g  = (const float*)d_in[4];
    const float* lnc_b  = (const float*)d_in[5];
    const float* Wq     = (const float*)d_in[6];
    const float* bq     = (const float*)d_in[7];
    const float* Wkv    = (const float*)d_in[8];
    const float* bkv    = (const float*)d_in[9];
    const float* Wo     = (const float*)d_in[10];
    const float* bo     = (const float*)d_in[11];

    float* out  = (float*)d_out;
    float* attn = out + (size_t)B_ * C_ * D_ * H_ * W_;   // second tuple output

    const int blocks = B_ * H_ * (W_ / PIX);              // 1024
    hipLaunchKernelGGL(vol_attn_kernel, dim3(blocks), dim3(256), 0, stream,
                       volume, feats, ln_g, ln_b, lnc_g, lnc_b,
                       Wq, bq, Wkv, bkv, Wo, bo, out, attn);
}